// DtNOperator_62577673502860
// MI455X (gfx1250) — compile-verified
//
#include <hip/hip_runtime.h>

// Problem constants (match reference)
#define GM    64          // grid size
#define GNB   252         // 4*GM-4 boundary points
#define ITERS 320         // Chebyshev iterations

typedef __attribute__((ext_vector_type(2))) float v2f;
typedef __attribute__((ext_vector_type(8))) float v8f;

// Dense difference matrix D (63x64 padded to 64x64), scaled by 1/h = 63.
// D[i][i] = -63, D[i][i+1] = +63, row 63 = 0.  Branchless -> v_cndmask,
// keeps EXEC uniform around WMMA.
__device__ __forceinline__ float dElem(int i, int j) {
  float v = (j == i) ? -63.0f : ((j == i + 1) ? 63.0f : 0.0f);
  return (i < 63) ? v : 0.0f;
}

// C(64x64) = [ACC ? C : 0] + A(64x64) * B(64x64), all row-major in LDS.
// AMODE/BMODE: 0 = from LDS pointer, 1 = analytic D, 2 = analytic D^T.
// 8 waves -> 16 tiles of 16x16, K stepped by 4 (V_WMMA_F32_16X16X4_F32).
// Fragment layouts per CDNA5 ISA 7.12.2:
//   A 16x4 : lanes 0-15 M=0..15 {v0=K0, v1=K1}; lanes 16-31 {v0=K2, v1=K3}
//   B 4x16 : mirrored (v0 = K0 / K2 rows, v1 = K1 / K3 rows, N = lane&15)
//   C 16x16: vgpr r: (M = mT + r + 8*(lane>>4), N = nT + (lane&15))
template <int AMODE, int BMODE, bool ACC>
__device__ __forceinline__ void mm64(const float* A, const float* B, float* C,
                                     int wave, int lane) {
  const int l15 = lane & 15;
  const int kh  = lane >> 4;  // 0 or 1: which K-half this lane holds
#pragma unroll
  for (int tt = 0; tt < 2; ++tt) {
    const int t    = wave + 8 * tt;      // tile id 0..15
    const int mT   = (t >> 2) << 4;
    const int nT   = (t & 3) << 4;
    const int crow = mT + 8 * kh;
    const int ccol = nT + l15;
    v8f c;
    if (ACC) {
#pragma unroll
      for (int r = 0; r < 8; ++r) c[r] = C[(crow + r) * 64 + ccol];
    } else {
#pragma unroll
      for (int r = 0; r < 8; ++r) c[r] = 0.0f;
    }
#pragma unroll
    for (int k0 = 0; k0 < 64; k0 += 4) {
      const int ka = k0 + 2 * kh;
      v2f af, bf;
      const int am = mT + l15;
      if (AMODE == 0)      { af[0] = A[am * 64 + ka]; af[1] = A[am * 64 + ka + 1]; }
      else if (AMODE == 1) { af[0] = dElem(am, ka);   af[1] = dElem(am, ka + 1);   }
      else                 { af[0] = dElem(ka, am);   af[1] = dElem(ka + 1, am);   }
      const int bn = nT + l15;
      if (BMODE == 0)      { bf[0] = B[ka * 64 + bn]; bf[1] = B[(ka + 1) * 64 + bn]; }
      else if (BMODE == 1) { bf[0] = dElem(ka, bn);   bf[1] = dElem(ka + 1, bn);     }
      else                 { bf[0] = dElem(bn, ka);   bf[1] = dElem(bn, ka + 1);     }
      // (neg_a, A, neg_b, B, c_mod, C, reuse_a, reuse_b)
      c = __builtin_amdgcn_wmma_f32_16x16x4_f32(false, af, false, bf, (short)0, c,
                                                false, false);
    }
#pragma unroll
    for (int r = 0; r < 8; ++r) C[(crow + r) * 64 + ccol] = c[r];
  }
}

// One workgroup per batch element. Whole solve lives in LDS (7*16KB = 112KB
// of the 320KB/WGP budget). 256 threads = 8 waves.
__global__ void __launch_bounds__(256)
dtn_cheby_wmma_kernel(const float* __restrict__ dbc,  // (32, 252)
                      const float* __restrict__ aG,   // (64, 64)
                      float* __restrict__ out) {      // (32, 252)
  __shared__ float sU[64 * 64];     // solution (boundary = Dirichlet data)
  __shared__ float sP[64 * 64];     // Chebyshev direction
  __shared__ float sV[64 * 64];     // matmul intermediate
  __shared__ float sLU[64 * 64];    // L*U
  __shared__ float sAx[64 * 64];    // 2*aL*aR/(aL+aR), x-faces (col 63 = 0)
  __shared__ float sAy[64 * 64];    // y-faces (row 63 = 0)
  __shared__ float sDinv[64 * 64];  // Jacobi 1/diag (0 on boundary)
  __shared__ float sSum;

  const int tid  = threadIdx.x;
  const int wave = tid >> 5;
  const int lane = tid & 31;
  const int b    = blockIdx.x;
  const float* g = dbc + b * GNB;

  // ---- setup: coefficients, Dirichlet embedding, zero state ----
  for (int e = tid; e < 64 * 64; e += 256) {
    const int y = e >> 6, x = e & 63;
    const float ac = aG[e];
    float ax = 0.0f, ay = 0.0f;
    if (x < 63) { const float ar = aG[e + 1];  ax = 2.0f * ac * ar / (ac + ar); }
    if (y < 63) { const float ad = aG[e + 64]; ay = 2.0f * ac * ad / (ac + ad); }
    sAx[e] = ax;
    sAy[e] = ay;
    // clockwise boundary embedding: top, right, bottom(rev), left(rev)
    float bv = 0.0f;
    if (y == 0 && x <= 62)       bv = g[x];
    else if (x == 63 && y <= 62) bv = g[63 + y];
    else if (y == 63 && x >= 1)  bv = g[189 - x];
    else if (x == 0 && y >= 1)   bv = g[252 - y];
    sU[e] = bv;
    sP[e] = 0.0f;
    sV[e] = 0.0f;
    sLU[e] = 0.0f;
  }
  __syncthreads();
  for (int e = tid; e < 64 * 64; e += 256) {
    const int y = e >> 6, x = e & 63;
    float dv = 0.0f;
    if (x >= 1 && x <= 62 && y >= 1 && y <= 62) {
      const float s = sAx[e - 1] + sAx[e] + sAy[e - 64] + sAy[e];
      dv = 1.0f / (s * 3969.0f);  // diag includes 1/h^2 = 63^2
    }
    sDinv[e] = dv;
  }
  __syncthreads();

  // ---- Chebyshev semi-iteration on D^-1 L, eig bounds [lmin, lmax] ----
  const float lmin = 1.0e-3f, lmax = 2.0f;
  const float theta = 0.5f * (lmax + lmin);
  const float delta = 0.5f * (lmax - lmin);
  const float sigma = theta / delta;
  float rho = 1.0f / sigma;

#pragma unroll 1
  for (int it = 0; it < ITERS; ++it) {
    // x-part: V = U * D^T ; V *= ax ; LU = V * D
    mm64<0, 2, false>(sU, sU, sV, wave, lane);
    __syncthreads();
    for (int e = tid; e < 64 * 64; e += 256) sV[e] *= sAx[e];
    __syncthreads();
    mm64<0, 1, false>(sV, sV, sLU, wave, lane);
    __syncthreads();
    // y-part: V = D * U ; V *= ay ; LU += D^T * V
    mm64<1, 0, false>(sU, sU, sV, wave, lane);
    __syncthreads();
    for (int e = tid; e < 64 * 64; e += 256) sV[e] *= sAy[e];
    __syncthreads();
    mm64<2, 0, true>(sV, sV, sLU, wave, lane);
    __syncthreads();
    // Chebyshev update: z = D^-1 (f - L u) = -Dinv*LU  (f interior = 0);
    // boundary: Dinv = 0 -> u fixed at Dirichlet data.
    if (it == 0) {
      const float c0 = 1.0f / theta;
      for (int e = tid; e < 64 * 64; e += 256) {
        const float p = c0 * (-sLU[e] * sDinv[e]);
        sP[e] = p;
        sU[e] += p;
      }
    } else {
      const float rho_new = 1.0f / (2.0f * sigma - rho);
      const float c1 = rho_new * rho;
      const float c2 = 2.0f * rho_new / delta;
      for (int e = tid; e < 64 * 64; e += 256) {
        const float z = -sLU[e] * sDinv[e];
        const float p = c1 * sP[e] + c2 * z;
        sP[e] = p;
        sU[e] += p;
      }
      rho = rho_new;
    }
    __syncthreads();
  }

  // ---- Neumann flux (clockwise, corners averaged), then mean-center ----
  if (tid == 0) sSum = 0.0f;
  __syncthreads();

  const float invh = 63.0f;
  float flux = 0.0f;
  const int q = tid;
  if (q < GNB) {
    if (q == 0) {
      flux = aG[0] * 0.5f * ((sU[0] - sU[64]) + (sU[0] - sU[1])) * invh;
    } else if (q <= 62) {
      const int j = q;
      flux = aG[j] * (sU[j] - sU[64 + j]) * invh;
    } else if (q == 63) {
      flux = aG[63] * 0.5f * ((sU[63] - sU[64 + 63]) + (sU[63] - sU[62])) * invh;
    } else if (q <= 125) {
      const int i = q - 63;
      flux = aG[i * 64 + 63] * (sU[i * 64 + 63] - sU[i * 64 + 62]) * invh;
    } else if (q == 126) {
      flux = aG[63 * 64 + 63] * 0.5f *
             ((sU[63 * 64 + 63] - sU[62 * 64 + 63]) +
              (sU[63 * 64 + 63] - sU[63 * 64 + 62])) * invh;
    } else if (q <= 188) {
      const int j = 189 - q;
      flux = aG[63 * 64 + j] * (sU[63 * 64 + j] - sU[62 * 64 + j]) * invh;
    } else if (q == 189) {
      flux = aG[63 * 64] * 0.5f *
             ((sU[63 * 64] - sU[62 * 64]) + (sU[63 * 64] - sU[63 * 64 + 1])) * invh;
    } else {
      const int i = 252 - q;
      flux = aG[i * 64] * (sU[i * 64] - sU[i * 64 + 1]) * invh;
    }
    atomicAdd(&sSum, flux);  // ds_add_f32
  }
  __syncthreads();
  const float mean = sSum * (1.0f / (float)GNB);
  if (q < GNB) out[b * GNB + q] = flux - mean;
}

extern "C" void kernel_launch(void* const* d_in, const int* in_sizes, int n_in,
                              void* d_out, int out_size, void* d_ws, size_t ws_size,
                              hipStream_t stream) {
  (void)in_sizes; (void)n_in; (void)d_ws; (void)ws_size; (void)out_size;
  const float* dbc = (const float*)d_in[0];  // (32, 252) float32
  const float* a   = (const float*)d_in[1];  // (64, 64)  float32
  float* out       = (float*)d_out;          // (32, 252) float32
  dtn_cheby_wmma_kernel<<<dim3(32), dim3(256), 0, stream>>>(dbc, a, out);
}